// Wasserstein3dPoseAdversary_56727928045820
// MI455X (gfx1250) — compile-verified
//
#include <hip/hip_runtime.h>
#include <math.h>

#define NPTS 4096
#define CHUNK 512
#define THREADS 256
#define ROWS_PER_BLOCK 128   // 8 waves * 16 rows

typedef __attribute__((ext_vector_type(2))) float v2f;
typedef __attribute__((ext_vector_type(8))) float v8f;
typedef __attribute__((ext_vector_type(4))) unsigned int v4u;
typedef __attribute__((ext_vector_type(8))) int v8i;
typedef __attribute__((ext_vector_type(4))) int v4i;

#if defined(__gfx1250__) && __has_builtin(__builtin_amdgcn_tensor_load_to_lds)
#define HAVE_TDM 1
#endif

// ---------------------------------------------------------------------------
// 1-D Tensor Data Mover copy: nelem f32 from gaddr to LDS at byte offset lds_off.
// D# per CDNA5 ISA ch.8: group0 {count=1, lds_addr, global_addr, type=2},
// group1 {data_size=4B, tensor_dim0=tile_dim0=nelem, tensor_dim1=1,
// tensor_dim0_stride=nelem}, remaining groups zero (<=2D tensor).
// This toolchain's builtin takes 6 args (g0 v4u, g1 v8i, v4i, v4i, v8i, cpol).
// ---------------------------------------------------------------------------
#ifdef HAVE_TDM
__device__ __forceinline__ void tdm_load_1d(unsigned lds_off, const void* gaddr,
                                            unsigned nelem) {
  unsigned long long ga = (unsigned long long)(uintptr_t)gaddr;
  v4u g0;
  g0.x = 1u;                                   // count=1 valid descriptor
  g0.y = lds_off;                              // lds_addr (bytes)
  g0.z = (unsigned)(ga & 0xFFFFFFFFu);         // global_addr[31:0]
  g0.w = (unsigned)((ga >> 32) & 0x01FFFFFFu)  // global_addr[56:32]
         | (2u << 30);                         // type=2 ("image")
  v8i g1;
  g1[0] = (int)(2u << 16);                     // data_size=2 (4 bytes)
  g1[1] = (int)((nelem & 0xFFFFu) << 16);      // tensor_dim0[15:0] @ [63:48]
  g1[2] = (int)((nelem >> 16) | (1u << 16));   // tensor_dim0[31:16], tensor_dim1=1
  g1[3] = (int)(nelem << 16);                  // tile_dim0 @ [127:112]
  g1[4] = 0;                                   // tile_dim1/2 unused
  g1[5] = (int)nelem;                          // tensor_dim0_stride[31:0]
  g1[6] = 0;
  g1[7] = 0;
  v4i z4 = {0, 0, 0, 0};
  v8i z8 = {0, 0, 0, 0, 0, 0, 0, 0};
  __builtin_amdgcn_tensor_load_to_lds(g0, g1, z4, z4, z8, 0);
}
#endif

// ---------------------------------------------------------------------------
// Operand packing.
// Translation: cost = 0.5*||x-y||^2 = X'.Y' with X'=[-x,0.5|x|^2,1,0...],
// Y'=[y,1,0.5|y|^2,0...], padded to K=8.
// Rotation: trace = sum_ij R1_ij*R2_ij, rows padded to K=12.
// ---------------------------------------------------------------------------
__global__ __launch_bounds__(THREADS)
void pack_t_kernel(const float* __restrict__ t, float* __restrict__ XP,
                   float* __restrict__ YP) {
  int i = blockIdx.x * blockDim.x + threadIdx.x;
  if (i >= NPTS) return;
  float x = t[3 * i + 0], y = t[3 * i + 1], z = t[3 * i + 2];
  float h = 0.5f * (x * x + y * y + z * z);
  float* xp = XP + (size_t)i * 8;
  xp[0] = -x; xp[1] = -y; xp[2] = -z; xp[3] = h; xp[4] = 1.f;
  xp[5] = 0.f; xp[6] = 0.f; xp[7] = 0.f;
  float* yp = YP + (size_t)i * 8;
  yp[0] = x; yp[1] = y; yp[2] = z; yp[3] = 1.f; yp[4] = h;
  yp[5] = 0.f; yp[6] = 0.f; yp[7] = 0.f;
}

__global__ __launch_bounds__(THREADS)
void pack_r_kernel(const float* __restrict__ R, float* __restrict__ P) {
  int i = blockIdx.x * blockDim.x + threadIdx.x;
  if (i >= NPTS) return;
  float* p = P + (size_t)i * 12;
#pragma unroll
  for (int k = 0; k < 9; ++k) p[k] = R[9 * i + k];
  p[9] = 0.f; p[10] = 0.f; p[11] = 0.f;
}

// ---------------------------------------------------------------------------
// Fused cost-tile (WMMA f32 16x16x4) + online logsumexp softmin.
// out[i] = -eps * log sum_j exp(h[j] - C_ij/eps),
//   h[j] = logw + (usePot ? pot[j]/eps : 0)
// avg: out[i] = 0.5*(prev[i] + softmin_i)
// SK = packed K (8 for translation, 12 for rotation), SK/4 WMMA K-steps.
// Y chunk staged into LDS by the Tensor Data Mover (wave 0 issues descriptor,
// all waves s_wait_tensorcnt + barrier); h staged by VALU concurrently.
// ---------------------------------------------------------------------------
template <int SK, bool ROT>
__global__ __launch_bounds__(THREADS)
void softmin_kernel(const float* __restrict__ X, const float* __restrict__ Y,
                    const float* __restrict__ pot, const float* __restrict__ prev,
                    float* __restrict__ out, float eps, float logw,
                    int avg, int usePot) {
  __shared__ __align__(128) float YS[CHUNK * SK];
  __shared__ __align__(128) float HS[CHUNK];

  const int tid  = threadIdx.x;
  const int lane = tid & 31;
  const int wave = tid >> 5;
  const int half = lane >> 4;   // 0: K pair {4s,4s+1}, 1: {4s+2,4s+3}
  const int l15  = lane & 15;
  const int rowBase = blockIdx.x * ROWS_PER_BLOCK + wave * 16;
  const float inv_eps = 1.0f / eps;

  // A fragment: 16 rows x SK, ISA 16x4 layout per K-step
  // (lanes 0-15: M=row,K={4s,4s+1}; lanes 16-31: M=row,K={4s+2,4s+3}).
  v2f a[SK / 4];
  {
    const int row = rowBase + l15;
#pragma unroll
    for (int s = 0; s < SK / 4; ++s) {
      const int k0 = 4 * s + 2 * half;
      a[s].x = X[(size_t)row * SK + k0];
      a[s].y = X[(size_t)row * SK + k0 + 1];
    }
  }

  // Online logsumexp state: D layout => lane holds column (l15) of 8 rows.
  float m[8], ssum[8];
#pragma unroll
  for (int r = 0; r < 8; ++r) { m[r] = -INFINITY; ssum[r] = 0.f; }

  for (int c0 = 0; c0 < NPTS; c0 += CHUNK) {
    __syncthreads();  // previous chunk fully consumed before overwrite
#ifdef HAVE_TDM
    if (wave == 0) {  // one wave issues the DMA descriptor
      tdm_load_1d((unsigned)(uintptr_t)&YS[0], Y + (size_t)c0 * SK, CHUNK * SK);
    }
#else
    for (int idx = tid; idx < CHUNK * SK; idx += THREADS)
      YS[idx] = Y[(size_t)c0 * SK + idx];
#endif
    // Stage h while the TDM streams Y.
    for (int j = tid; j < CHUNK; j += THREADS)
      HS[j] = logw + (usePot ? pot[c0 + j] * inv_eps : 0.f);
    // Prefetch next chunk toward L2 (global_prefetch_b8).
    if (c0 + CHUNK < NPTS)
      __builtin_prefetch(&Y[(size_t)(c0 + CHUNK) * SK + tid * 16], 0, 1);
#ifdef HAVE_TDM
    __builtin_amdgcn_s_wait_tensorcnt(0);  // no-op for waves with TENSORcnt==0
#endif
    __syncthreads();

#pragma unroll 1
    for (int jt = 0; jt < CHUNK / 16; ++jt) {
      v8f acc = {};
      const int col = jt * 16 + l15;
#pragma unroll
      for (int s = 0; s < SK / 4; ++s) {
        v2f b;
        const int k0 = 4 * s + 2 * half;
        b.x = YS[col * SK + k0];
        b.y = YS[col * SK + k0 + 1];
        acc = __builtin_amdgcn_wmma_f32_16x16x4_f32(
            false, a[s], false, b, (short)0, acc, false, false);
      }
      const float hj = HS[jt * 16 + l15];
#pragma unroll
      for (int r = 0; r < 8; ++r) {
        float dot = acc[r];
        float cost;
        if (ROT) {
          float c = (dot - 1.0f) * 0.5f;
          c = fminf(fmaxf(c, -1.0f + 1e-6f), 1.0f - 1e-6f);
          cost = acosf(c);
        } else {
          cost = dot;
        }
        const float v  = hj - cost * inv_eps;
        const float mn = fmaxf(m[r], v);
        ssum[r] = ssum[r] * __expf(m[r] - mn) + __expf(v - mn);
        m[r] = mn;
      }
    }
  }

  // Reduce across the 16 lanes of each half-wave (same rows, diff columns).
#pragma unroll
  for (int r = 0; r < 8; ++r) {
    float mm = m[r], ss = ssum[r];
#pragma unroll
    for (int d = 1; d < 16; d <<= 1) {
      const float mo = __shfl_xor(mm, d, 32);
      const float so = __shfl_xor(ss, d, 32);
      const float mn = fmaxf(mm, mo);
      ss = ss * __expf(mm - mn) + so * __expf(mo - mn);
      mm = mn;
    }
    if (l15 == 0) {
      const int row = rowBase + r + 8 * half;
      const float val = -eps * (mm + __logf(ss));
      out[row] = avg ? 0.5f * (prev[row] + val) : val;
    }
  }
}

// ---------------------------------------------------------------------------
// Final reductions.
// ---------------------------------------------------------------------------
__global__ __launch_bounds__(THREADS)
void loss_reduce_kernel(const float* __restrict__ fba, const float* __restrict__ faa,
                        const float* __restrict__ gab, const float* __restrict__ gbb,
                        float* __restrict__ out) {
  __shared__ float smem[THREADS];
  float acc = 0.f;
  for (int i = threadIdx.x; i < NPTS; i += THREADS)
    acc += (fba[i] - faa[i]) + (gab[i] - gbb[i]);
  smem[threadIdx.x] = acc;
  __syncthreads();
  for (int s = THREADS / 2; s > 0; s >>= 1) {
    if (threadIdx.x < s) smem[threadIdx.x] += smem[threadIdx.x + s];
    __syncthreads();
  }
  if (threadIdx.x == 0) out[0] = smem[0] * (1.0f / NPTS);
}

__global__ void combine_kernel(const float* sR, const float* sT, float* out) {
  if (threadIdx.x == 0) out[0] = sR[0] + 3.14159265358979323846f * sT[0];
}

// ---------------------------------------------------------------------------
// Host orchestration.
// ---------------------------------------------------------------------------
static void launch_softmin(bool rot, const float* X, const float* Y,
                           const float* pot, const float* prev, float* out,
                           float eps, float logw, int avg, int usePot,
                           hipStream_t s) {
  dim3 g(NPTS / ROWS_PER_BLOCK), b(THREADS);
  if (rot)
    softmin_kernel<12, true><<<g, b, 0, s>>>(X, Y, pot, prev, out, eps, logw, avg, usePot);
  else
    softmin_kernel<8, false><<<g, b, 0, s>>>(X, Y, pot, prev, out, eps, logw, avg, usePot);
}

static int build_eps(double diam, double* eps) {
  int n = 0;
  eps[n++] = diam * diam;
  const double stop = 2.0 * log(0.05), step = 2.0 * log(0.5);
  for (double e = 2.0 * log(diam); e > stop; e += step) eps[n++] = exp(e);
  eps[n++] = 0.05 * 0.05;
  return n;
}

static void sinkhorn(bool rot,
                     const float* Xa, const float* Ya,   // buffer-set operands (X-role / Y-role)
                     const float* Xb, const float* Yb,   // fake-set operands
                     float* fba0, float* fba1, float* gab0, float* gab1,
                     float* faa0, float* faa1, float* gbb0, float* gbb1,
                     double diam, float logw, float* out_scalar, hipStream_t s) {
  double eps_list[16];
  const int neps = build_eps(diam, eps_list);
  float* fba[2] = {fba0, fba1};
  float* gab[2] = {gab0, gab1};
  float* faa[2] = {faa0, faa1};
  float* gbb[2] = {gbb0, gbb1};
  int c = 0;
  const float eps0 = (float)eps_list[0];

  // Init at eps0 (no potentials).
  launch_softmin(rot, Xa, Ya, faa[c], faa[c], faa[c], eps0, logw, 0, 0, s);  // C_xx, a_log
  launch_softmin(rot, Xb, Yb, gbb[c], gbb[c], gbb[c], eps0, logw, 0, 0, s);  // C_yy, b_log
  launch_softmin(rot, Xb, Ya, gab[c], gab[c], gab[c], eps0, logw, 0, 0, s);  // C_yx, a_log
  launch_softmin(rot, Xa, Yb, fba[c], fba[c], fba[c], eps0, logw, 0, 0, s);  // C_xy, b_log

  for (int it = 0; it < neps; ++it) {
    const float eps = (float)eps_list[it];
    const int a = 1 - c;
    launch_softmin(rot, Xa, Yb, gab[c], fba[c], fba[a], eps, logw, 1, 1, s);  // ft -> fba'
    launch_softmin(rot, Xb, Ya, fba[c], gab[c], gab[a], eps, logw, 1, 1, s);  // gt -> gab'
    launch_softmin(rot, Xa, Ya, faa[c], faa[c], faa[a], eps, logw, 1, 1, s);
    launch_softmin(rot, Xb, Yb, gbb[c], gbb[c], gbb[a], eps, logw, 1, 1, s);
    c = a;
  }

  // Final non-averaged extrapolation at eps = blur^2.
  {
    const float eps = (float)eps_list[neps - 1];
    const int a = 1 - c;
    launch_softmin(rot, Xa, Yb, gab[c], fba[a], fba[a], eps, logw, 0, 1, s);
    launch_softmin(rot, Xb, Ya, fba[c], gab[a], gab[a], eps, logw, 0, 1, s);
    launch_softmin(rot, Xa, Ya, faa[c], faa[a], faa[a], eps, logw, 0, 1, s);
    launch_softmin(rot, Xb, Yb, gbb[c], gbb[a], gbb[a], eps, logw, 0, 1, s);
    c = a;
  }

  loss_reduce_kernel<<<1, THREADS, 0, s>>>(fba[c], faa[c], gab[c], gbb[c], out_scalar);
}

extern "C" void kernel_launch(void* const* d_in, const int* in_sizes, int n_in,
                              void* d_out, int out_size, void* d_ws, size_t ws_size,
                              hipStream_t stream) {
  const float* r_buffer = (const float*)d_in[0];  // [4096,3,3]
  const float* t_buffer = (const float*)d_in[1];  // [4096,3]
  const float* R_fake   = (const float*)d_in[2];  // [4096,3,3]
  const float* t_fake   = (const float*)d_in[3];  // [4096,3]
  float* out = (float*)d_out;

  float* ws = (float*)d_ws;
  size_t o = 0;
  float* TXb = ws + o; o += (size_t)NPTS * 8;
  float* TYb = ws + o; o += (size_t)NPTS * 8;
  float* TXf = ws + o; o += (size_t)NPTS * 8;
  float* TYf = ws + o; o += (size_t)NPTS * 8;
  float* Rb  = ws + o; o += (size_t)NPTS * 12;
  float* Rf  = ws + o; o += (size_t)NPTS * 12;
  float* pot = ws + o; o += (size_t)16 * NPTS;  // 2 losses * 4 potentials * 2 buffers
  float* sT  = ws + o; o += 1;
  float* sR  = ws + o; o += 1;
  (void)ws_size; (void)in_sizes; (void)n_in; (void)out_size;

  const dim3 pg(NPTS / THREADS), pb(THREADS);
  pack_t_kernel<<<pg, pb, 0, stream>>>(t_buffer, TXb, TYb);
  pack_t_kernel<<<pg, pb, 0, stream>>>(t_fake, TXf, TYf);
  pack_r_kernel<<<pg, pb, 0, stream>>>(r_buffer, Rb);
  pack_r_kernel<<<pg, pb, 0, stream>>>(R_fake, Rf);

  const float logw = logf(1.0f / (float)NPTS);
  float* P = pot;  // 16 buffers of NPTS floats

  // Translation loss (cost = 0.5*sqdist via augmented K=8 dot), diameter 2.0.
  sinkhorn(false, TXb, TYb, TXf, TYf,
           P + 0 * NPTS, P + 1 * NPTS, P + 2 * NPTS, P + 3 * NPTS,
           P + 4 * NPTS, P + 5 * NPTS, P + 6 * NPTS, P + 7 * NPTS,
           2.0, logw, sT, stream);

  // Rotation loss (cost = arccos((tr-1)/2) via K=12 dot), diameter 3.141.
  sinkhorn(true, Rb, Rb, Rf, Rf,
           P + 8 * NPTS, P + 9 * NPTS, P + 10 * NPTS, P + 11 * NPTS,
           P + 12 * NPTS, P + 13 * NPTS, P + 14 * NPTS, P + 15 * NPTS,
           3.141, logw, sR, stream);

  combine_kernel<<<1, 32, 0, stream>>>(sR, sT, out);
}